// GRU_ATTENTIONDecoder_83373905150538
// MI455X (gfx1250) — compile-verified
//
#include <hip/hip_runtime.h>
#include <hip/hip_bf16.h>

#define HDIM 1024
#define BDIM 64
#define SDIM 1024
#define VDIM 32000

typedef __attribute__((ext_vector_type(2))) float v2f;
typedef __attribute__((ext_vector_type(4))) float v4f;
typedef __attribute__((ext_vector_type(8))) float v8f;

__device__ __forceinline__ float fast_tanh(float x) {
  // tanh(x) = 1 - 2/(1+exp(2x)); exp(2x) = exp2(2*log2(e)*x)
  float e = __builtin_amdgcn_exp2f(x * 2.88539008177792681f);
  return 1.0f - 2.0f * __builtin_amdgcn_rcpf(1.0f + e);
}
__device__ __forceinline__ float fast_sigmoid(float x) {
  return __builtin_amdgcn_rcpf(1.0f + __builtin_amdgcn_exp2f(-1.44269504088896341f * x));
}
__device__ __forceinline__ float fast_exp(float x) {
  return __builtin_amdgcn_exp2f(1.44269504088896341f * x);
}

// ---------------------------------------------------------------------------
// 1) c_b = sum_h tanh(h[b,h]) * W_energy[0,h]  + b_energy   (S-invariant term)
// ---------------------------------------------------------------------------
__global__ void cb_kernel(const float* __restrict__ h, const float* __restrict__ W_energy,
                          const float* __restrict__ b_energy, float* __restrict__ cb) {
  __shared__ float red[256];
  int b = blockIdx.x;
  float s = 0.f;
  for (int i = threadIdx.x; i < HDIM; i += 256)
    s += fast_tanh(h[b * HDIM + i]) * W_energy[i];
  red[threadIdx.x] = s;
  __syncthreads();
  for (int off = 128; off > 0; off >>= 1) {
    if (threadIdx.x < (unsigned)off) red[threadIdx.x] += red[threadIdx.x + off];
    __syncthreads();
  }
  if (threadIdx.x == 0) cb[b] = red[0] + b_energy[0];
}

// ---------------------------------------------------------------------------
// 2) energy[s,b] = relu(c_b + sum_h tanh(enc[s,b,h]) * w_e[h])
//    one wave32 per (s,b) row; enc is streamed non-temporally (256 MB)
// ---------------------------------------------------------------------------
__global__ void energy_kernel(const float* __restrict__ enc, const float* __restrict__ we,
                              const float* __restrict__ cb, float* __restrict__ energy) {
  int row  = blockIdx.x * 8 + (threadIdx.x >> 5);   // row = s*B + b
  int lane = threadIdx.x & 31;
  const v4f* p  = (const v4f*)(enc + (size_t)row * HDIM);
  const v4f* wv = (const v4f*)we;
  float sum = 0.f;
#pragma unroll
  for (int i = 0; i < 8; ++i) {
    v4f v = __builtin_nontemporal_load(p + i * 32 + lane);
    v4f w = wv[i * 32 + lane];
    sum += fast_tanh(v.x) * w.x + fast_tanh(v.y) * w.y +
           fast_tanh(v.z) * w.z + fast_tanh(v.w) * w.w;
  }
  for (int off = 16; off > 0; off >>= 1) sum += __shfl_xor(sum, off, 32);
  if (lane == 0) {
    float e = cb[row & (BDIM - 1)] + sum;
    energy[row] = e > 0.f ? e : 0.f;
  }
}

// ---------------------------------------------------------------------------
// 3) column softmax over S for each b
// ---------------------------------------------------------------------------
__global__ void softmax_kernel(const float* __restrict__ energy, float* __restrict__ attn) {
  __shared__ float red[256];
  int b = blockIdx.x;
  int t = threadIdx.x;
  float m = -3.0e38f;
  for (int s = t; s < SDIM; s += 256) m = fmaxf(m, energy[s * BDIM + b]);
  red[t] = m; __syncthreads();
  for (int off = 128; off > 0; off >>= 1) {
    if (t < off) red[t] = fmaxf(red[t], red[t + off]);
    __syncthreads();
  }
  m = red[0]; __syncthreads();
  float sum = 0.f;
  for (int s = t; s < SDIM; s += 256) sum += fast_exp(energy[s * BDIM + b] - m);
  red[t] = sum; __syncthreads();
  for (int off = 128; off > 0; off >>= 1) {
    if (t < off) red[t] += red[t + off];
    __syncthreads();
  }
  float inv = __builtin_amdgcn_rcpf(red[0]);
  for (int s = t; s < SDIM; s += 256)
    attn[s * BDIM + b] = fast_exp(energy[s * BDIM + b] - m) * inv;
}

// ---------------------------------------------------------------------------
// 4) context[b,h] = sum_s attn[s,b] * enc[s,b,h] -> xbuf[b, 0:1024]
// ---------------------------------------------------------------------------
__global__ void context_kernel(const float* __restrict__ enc, const float* __restrict__ attn,
                               float* __restrict__ xbuf) {
  int b  = blockIdx.x >> 2;
  int hh = ((blockIdx.x & 3) << 8) + threadIdx.x;
  float acc = 0.f;
#pragma unroll 8
  for (int s = 0; s < SDIM; ++s)
    acc += attn[s * BDIM + b] *
           __builtin_nontemporal_load(enc + ((size_t)s * BDIM + b) * HDIM + hh);
  xbuf[b * 2 * HDIM + hh] = acc;
}

// ---------------------------------------------------------------------------
// 5) xbuf[b, 1024:2048] = emb_table[decoding_input[b], :]
// ---------------------------------------------------------------------------
__global__ void embed_kernel(const int* __restrict__ idx, const float* __restrict__ emb,
                             float* __restrict__ xbuf) {
  int b = blockIdx.x;
  int r = idx[b];
  v4f v = ((const v4f*)(emb + (size_t)r * HDIM))[threadIdx.x];
  ((v4f*)(xbuf + b * 2 * HDIM + HDIM))[threadIdx.x] = v;
}

// ---------------------------------------------------------------------------
// 6/7/9) C(64xN) = A(64xK) @ W(NxK)^T + bias, fp32 WMMA 16x16x4.
//   One wave per 16-wide N tile; 4 M-tile accumulators cover all 64 rows so
//   each weight element is streamed from HBM exactly once (non-temporal).
//   Explicit depth-2 software pipeline with disjoint register slots and
//   sched_barrier fences so prefetch loads stay a full iteration ahead of
//   the WMMAs that consume them.
//   A frag: lane%16 = M row, k = kbase + 2*(lane/16) + {0,1}
//   B frag: lane%16 = N col (a row of W),  same k split
//   D:      lane%16 = N col, m = vgpr + 8*(lane/16)
// ---------------------------------------------------------------------------
#define LDSLOT(dB, d0, d1, d2, d3, kk)                          \
  dB = __builtin_nontemporal_load((const v2f*)(wp + (kk)));     \
  d0 = *(const v2f*)(ap + (kk));                                \
  d1 = *(const v2f*)(ap + 16 * K + (kk));                       \
  d2 = *(const v2f*)(ap + 32 * K + (kk));                       \
  d3 = *(const v2f*)(ap + 48 * K + (kk));

#define DOWMMA(xB, x0, x1, x2, x3)                                                                     \
  acc0 = __builtin_amdgcn_wmma_f32_16x16x4_f32(false, x0, false, xB, (short)0, acc0, false, false);    \
  acc1 = __builtin_amdgcn_wmma_f32_16x16x4_f32(false, x1, false, xB, (short)0, acc1, false, false);    \
  acc2 = __builtin_amdgcn_wmma_f32_16x16x4_f32(false, x2, false, xB, (short)0, acc2, false, false);    \
  acc3 = __builtin_amdgcn_wmma_f32_16x16x4_f32(false, x3, false, xB, (short)0, acc3, false, false);

__global__ void gemm64_wmma_kernel(const float* __restrict__ A,
                                   const float* __restrict__ W,
                                   const float* __restrict__ bias,
                                   float* __restrict__ C,
                                   int K, int N) {
  const int lane  = threadIdx.x & 31;
  const int wave  = threadIdx.x >> 5;
  const int ntile = blockIdx.x * (blockDim.x >> 5) + wave;
  if (ntile * 16 >= N) return;               // wave-uniform: EXEC stays all-ones
  const int nl    = lane & 15;
  const int khalf = (lane >> 4) << 1;        // 0 or 2
  const int n     = ntile * 16 + nl;
  const float* wp = W + (size_t)n * K + khalf;
  const float* ap = A + (size_t)nl * K + khalf;
  v8f acc0 = {}, acc1 = {}, acc2 = {}, acc3 = {};

  v2f b0, a00, a01, a02, a03;   // slot 0
  v2f b1, a10, a11, a12, a13;   // slot 1
  LDSLOT(b0, a00, a01, a02, a03, 0);
  LDSLOT(b1, a10, a11, a12, a13, 4);

  for (int k = 8; k < K; k += 8) {
    v2f nb0, n00, n01, n02, n03;
    LDSLOT(nb0, n00, n01, n02, n03, k);
    __builtin_amdgcn_sched_barrier(0);
    DOWMMA(b0, a00, a01, a02, a03);
    b0 = nb0; a00 = n00; a01 = n01; a02 = n02; a03 = n03;

    v2f nb1, n10, n11, n12, n13;
    LDSLOT(nb1, n10, n11, n12, n13, k + 4);
    __builtin_amdgcn_sched_barrier(0);
    DOWMMA(b1, a10, a11, a12, a13);
    b1 = nb1; a10 = n10; a11 = n11; a12 = n12; a13 = n13;
  }
  DOWMMA(b0, a00, a01, a02, a03);
  DOWMMA(b1, a10, a11, a12, a13);

  float bv  = bias[n];
  int mrow  = (lane >> 4) * 8;
#pragma unroll
  for (int v = 0; v < 8; ++v) {
    C[(size_t)(     mrow + v) * N + n] = acc0[v] + bv;
    C[(size_t)(16 + mrow + v) * N + n] = acc1[v] + bv;
    C[(size_t)(32 + mrow + v) * N + n] = acc2[v] + bv;
    C[(size_t)(48 + mrow + v) * N + n] = acc3[v] + bv;
  }
}

// ---------------------------------------------------------------------------
// 8) GRU gates -> h_new (to ws and to output slot)
// ---------------------------------------------------------------------------
__global__ void gates_kernel(const float* __restrict__ gi, const float* __restrict__ gh,
                             const float* __restrict__ h, float* __restrict__ hnew,
                             float* __restrict__ out_h) {
  int i = blockIdx.x * 256 + threadIdx.x;     // 0 .. B*H-1
  int b = i >> 10;
  int j = i & (HDIM - 1);
  const float* gib = gi + b * 3 * HDIM;
  const float* ghb = gh + b * 3 * HDIM;
  float r = fast_sigmoid(gib[j]            + ghb[j]);
  float z = fast_sigmoid(gib[HDIM + j]     + ghb[HDIM + j]);
  float n = fast_tanh  (gib[2 * HDIM + j] + r * ghb[2 * HDIM + j]);
  float hv = h[i];
  float o  = (1.f - z) * n + z * hv;
  hnew[i]  = o;
  out_h[i] = o;
}

extern "C" void kernel_launch(void* const* d_in, const int* in_sizes, int n_in,
                              void* d_out, int out_size, void* d_ws, size_t ws_size,
                              hipStream_t stream) {
  const int*   dec_in   = (const int*)  d_in[0];
  const float* dec_h    = (const float*)d_in[1];   // (1,B,H)
  const float* enc      = (const float*)d_in[2];   // (S,B,H)
  const float* emb      = (const float*)d_in[3];   // (V,E)
  const float* W_ih     = (const float*)d_in[4];   // (3H,2E)
  const float* b_ih     = (const float*)d_in[5];
  const float* W_hh     = (const float*)d_in[6];   // (3H,H)
  const float* b_hh     = (const float*)d_in[7];
  const float* W_energy = (const float*)d_in[8];   // (1,2H)
  const float* b_energy = (const float*)d_in[9];
  const float* W_out    = (const float*)d_in[10];  // (V,H)
  const float* b_out    = (const float*)d_in[11];

  float* out_logits = (float*)d_out;                        // B*V
  float* out_h      = out_logits + (size_t)BDIM * VDIM;     // B*H

  float* ws     = (float*)d_ws;
  float* cb     = ws;                          // 64
  float* energy = cb + 64;                     // S*B
  float* attn   = energy + SDIM * BDIM;        // S*B
  float* xbuf   = attn + SDIM * BDIM;          // B*2H
  float* gi     = xbuf + BDIM * 2 * HDIM;      // B*3H
  float* gh     = gi + BDIM * 3 * HDIM;        // B*3H
  float* hnew   = gh + BDIM * 3 * HDIM;        // B*H

  cb_kernel     <<<BDIM,              256, 0, stream>>>(dec_h, W_energy, b_energy, cb);
  energy_kernel <<<SDIM * BDIM / 8,   256, 0, stream>>>(enc, W_energy + HDIM, cb, energy);
  softmax_kernel<<<BDIM,              256, 0, stream>>>(energy, attn);
  context_kernel<<<BDIM * (HDIM/256), 256, 0, stream>>>(enc, attn, xbuf);
  embed_kernel  <<<BDIM,              256, 0, stream>>>(dec_in, emb, xbuf);
  gemm64_wmma_kernel<<<(3 * HDIM / 16) / 8, 256, 0, stream>>>(xbuf,  W_ih, b_ih, gi, 2 * HDIM, 3 * HDIM);
  gemm64_wmma_kernel<<<(3 * HDIM / 16) / 8, 256, 0, stream>>>(dec_h, W_hh, b_hh, gh, HDIM,     3 * HDIM);
  gates_kernel  <<<BDIM * HDIM / 256, 256, 0, stream>>>(gi, gh, dec_h, hnew, out_h);
  gemm64_wmma_kernel<<<(VDIM / 16) / 8, 256, 0, stream>>>(hnew, W_out, b_out, out_logits, HDIM, VDIM);
}